// GraphConvolutionalNetwork_50895362457878
// MI455X (gfx1250) — compile-verified
//
#include <hip/hip_runtime.h>

#define N_NODES 100000
#define N_EDGES 3200000
#define N_FEAT  512
#define HIDDEN  16

typedef __attribute__((ext_vector_type(2))) float v2f;
typedef __attribute__((ext_vector_type(8))) float v8f;

// ---------------------------------------------------------------------------
// Kernel 1: H1 = x @ W1 using V_WMMA_F32_16X16X4_F32.
// One wave (32 lanes) computes one 16x16 output tile (16 node rows x 16 hidden).
// 100000 rows = 6250 exact tiles -> whole waves retire uniformly, EXEC stays
// all-1s across every WMMA as the ISA requires.
// A layout (32-bit 16x4): lanes 0-15 hold row M=lane, K={k0,k0+1};
// lanes 16-31 hold row M=lane-16, K={k0+2,k0+3}  -> one b64 load per lane.
// B staged in LDS with pitch 17 (conflict-free, fuses into ds_load_2addr_b32).
// ---------------------------------------------------------------------------
#define W1_PITCH 17

__global__ __launch_bounds__(256) void gcn_gemm1_wmma(
    const float* __restrict__ x, const float* __restrict__ W1,
    float* __restrict__ h1, int ntiles) {
  __shared__ float w1s[N_FEAT * W1_PITCH];
  const int tid = threadIdx.x;
  // Stage W1 [512,16] into LDS, swizzled pitch to kill bank conflicts.
  for (int i = tid; i < N_FEAT * HIDDEN; i += 256) {
    int k = i >> 4, n = i & 15;
    w1s[k * W1_PITCH + n] = W1[i];
  }
  __syncthreads();

  const int wave = tid >> 5;
  const int lane = tid & 31;
  const int half = lane >> 4;   // 0: lanes 0-15, 1: lanes 16-31
  const int mrow = lane & 15;
  const int tile = blockIdx.x * 8 + wave;
  if (tile >= ntiles) return;   // uniform per wave

  const float* __restrict__ xrow =
      x + (size_t)(tile * 16 + mrow) * N_FEAT + 2 * half;

  v8f acc = {};
#pragma unroll 4
  for (int k0 = 0; k0 < N_FEAT; k0 += 4) {
    v2f a = *(const v2f*)(xrow + k0);                       // global_load_b64
    v2f b;
    b.x = w1s[(k0 + 2 * half) * W1_PITCH + mrow];           // ds_load_2addr
    b.y = w1s[(k0 + 2 * half + 1) * W1_PITCH + mrow];
    acc = __builtin_amdgcn_wmma_f32_16x16x4_f32(
        /*neg_a=*/false, a, /*neg_b=*/false, b,
        /*c_mod=*/(short)0, acc, /*reuse_a=*/false, /*reuse_b=*/false);
  }

  // C/D layout: VGPR j -> row M = j + 8*half, col N = mrow.
  float* __restrict__ out = h1 + (size_t)(tile * 16 + 8 * half) * HIDDEN + mrow;
#pragma unroll
  for (int j = 0; j < 8; ++j) out[(size_t)j * HIDDEN] = acc[j];
}

// ---------------------------------------------------------------------------
// Kernel 2: zero the aggregation buffers (graph-capture safe; no memset).
// ---------------------------------------------------------------------------
__global__ void gcn_zero(float* __restrict__ p, int n) {
  int i = blockIdx.x * blockDim.x + threadIdx.x;
  for (; i < n; i += gridDim.x * blockDim.x) p[i] = 0.0f;
}

// ---------------------------------------------------------------------------
// Kernel 3: SpMM layer 1.  thread = (edge, feature); 16 lanes share one edge,
// so the gather h1[col*16+f] and the scatter agg1[row*16+f] are 64B-contiguous
// per edge-group.  Both tables are 6.4MB -> L2 resident.  unsafeAtomicAdd
// guarantees the native CDNA5 GLOBAL_ATOMIC_ADD_F32 (no cmpswap retry loop)
// on the runtime-dominant scatter path.
// ---------------------------------------------------------------------------
__global__ __launch_bounds__(256) void gcn_spmm1(
    const float* __restrict__ vals, const int* __restrict__ rows,
    const int* __restrict__ cols, const float* __restrict__ h1,
    float* __restrict__ agg1) {
  int idx = blockIdx.x * blockDim.x + threadIdx.x;
  int e = idx >> 4;
  int f = idx & 15;
  if (e >= N_EDGES) return;
  int r = rows[e];
  int c = cols[e];
  float v = vals[e];
  float contrib = v * h1[(size_t)c * HIDDEN + f];
  unsafeAtomicAdd(&agg1[(size_t)r * HIDDEN + f], contrib);
}

// ---------------------------------------------------------------------------
// Kernel 4: fused bias + relu + (h @ W2) per node -> g[n]  (HIDDEN=16 -> OUT=1)
// ---------------------------------------------------------------------------
__global__ __launch_bounds__(256) void gcn_layer2_dense(
    const float* __restrict__ agg1, const float* __restrict__ b1,
    const float* __restrict__ W2, float* __restrict__ g) {
  int n = blockIdx.x * blockDim.x + threadIdx.x;
  if (n >= N_NODES) return;
  const float4* __restrict__ row = (const float4*)(agg1 + (size_t)n * HIDDEN);
  float s = 0.0f;
#pragma unroll
  for (int q = 0; q < 4; ++q) {
    float4 v = row[q];
    s += fmaxf(v.x + b1[4 * q + 0], 0.0f) * W2[4 * q + 0];
    s += fmaxf(v.y + b1[4 * q + 1], 0.0f) * W2[4 * q + 1];
    s += fmaxf(v.z + b1[4 * q + 2], 0.0f) * W2[4 * q + 2];
    s += fmaxf(v.w + b1[4 * q + 3], 0.0f) * W2[4 * q + 3];
  }
  g[n] = s;
}

// ---------------------------------------------------------------------------
// Kernel 5: SpMM layer 2 (scalar): agg2[row] += val * g[col]
// ---------------------------------------------------------------------------
__global__ __launch_bounds__(256) void gcn_spmm2(
    const float* __restrict__ vals, const int* __restrict__ rows,
    const int* __restrict__ cols, const float* __restrict__ g,
    float* __restrict__ agg2) {
  int e = blockIdx.x * blockDim.x + threadIdx.x;
  if (e >= N_EDGES) return;
  float contrib = vals[e] * g[cols[e]];
  unsafeAtomicAdd(&agg2[rows[e]], contrib);
}

// ---------------------------------------------------------------------------
// Kernel 6: out = sigmoid(agg2 + b2)
// ---------------------------------------------------------------------------
__global__ __launch_bounds__(256) void gcn_sigmoid(
    const float* __restrict__ agg2, const float* __restrict__ b2,
    float* __restrict__ out) {
  int n = blockIdx.x * blockDim.x + threadIdx.x;
  if (n >= N_NODES) return;
  float z = agg2[n] + b2[0];
  out[n] = 1.0f / (1.0f + __expf(-z));
}

// ---------------------------------------------------------------------------
extern "C" void kernel_launch(void* const* d_in, const int* in_sizes, int n_in,
                              void* d_out, int out_size, void* d_ws,
                              size_t ws_size, hipStream_t stream) {
  const float* x       = (const float*)d_in[0];  // [100000,512]
  const float* lapvals = (const float*)d_in[1];  // [3.2M]
  const float* W1      = (const float*)d_in[2];  // [512,16]
  const float* b1      = (const float*)d_in[3];  // [16]
  const float* W2      = (const float*)d_in[4];  // [16,1]
  const float* b2      = (const float*)d_in[5];  // [1]
  const int*   laprows = (const int*)d_in[6];    // [3.2M]
  const int*   lapcols = (const int*)d_in[7];    // [3.2M]
  float* out = (float*)d_out;                    // [100000,1]

  float* ws   = (float*)d_ws;
  float* h1   = ws;                 // 100000*16
  float* agg1 = ws + 1600000;       // 100000*16
  float* g    = ws + 3200000;       // 100000
  float* agg2 = ws + 3300000;       // 100000
  (void)in_sizes; (void)n_in; (void)out_size; (void)ws_size;

  const int NTILES = N_NODES / 16;  // 6250, exact

  // Layer-1 GEMM (WMMA fp32) — independent of aggregation zeroing order.
  gcn_gemm1_wmma<<<(NTILES + 7) / 8, 256, 0, stream>>>(x, W1, h1, NTILES);

  // Zero both aggregation buffers (1.6M + 0.1M floats).
  gcn_zero<<<2048, 256, 0, stream>>>(agg1, N_NODES * HIDDEN);
  gcn_zero<<<512, 256, 0, stream>>>(agg2, N_NODES);

  // Layer-1 sparse aggregation: 3.2M edges * 16 features.
  {
    long total = (long)N_EDGES * HIDDEN;  // 51.2M threads
    int blocks = (int)((total + 255) / 256);
    gcn_spmm1<<<blocks, 256, 0, stream>>>(lapvals, laprows, lapcols, h1, agg1);
  }

  // Fused relu+bias + dense layer 2.
  gcn_layer2_dense<<<(N_NODES + 255) / 256, 256, 0, stream>>>(agg1, b1, W2, g);

  // Layer-2 sparse aggregation.
  gcn_spmm2<<<(N_EDGES + 255) / 256, 256, 0, stream>>>(lapvals, laprows,
                                                       lapcols, g, agg2);

  // Fused bias + sigmoid.
  gcn_sigmoid<<<(N_NODES + 255) / 256, 256, 0, stream>>>(agg2, b2, out);
}